// Dist_Conv2D_Dense_90855738180333
// MI455X (gfx1250) — compile-verified
//
#include <hip/hip_runtime.h>
#include <hip/hip_bf16.h>

// Problem constants (match reference setup_inputs)
#define B_    8
#define CIN   16
#define HH    64
#define WW    64
#define KK    3
#define COUT  32
#define D_    (CIN * KK * KK)   // 144
#define TW    16                // tile width  (pixels)
#define TH    8                 // tile height (pixels)
#define XTW   (TW + KK - 1)     // 18 (halo)
#define XTH   (TH + KK - 1)     // 10 (halo)
#define NTHREADS 128

#define AS1 __attribute__((address_space(1)))
#define AS3 __attribute__((address_space(3)))

// ---- CDNA5 async global->LDS copy (ASYNCcnt-tracked), with safe fallback ----
__device__ __forceinline__ void async_ld_b32(const float* __restrict__ g, float* l) {
#if __has_builtin(__builtin_amdgcn_global_load_async_to_lds_b32)
  __builtin_amdgcn_global_load_async_to_lds_b32(
      (AS1 int*)g, (AS3 int*)l, /*offset=*/0, /*cpol=*/0);
#else
  *l = *g;   // plain global_load + ds_store fallback
#endif
}

__device__ __forceinline__ void wait_async0() {
#if __has_builtin(__builtin_amdgcn_s_wait_asynccnt)
  __builtin_amdgcn_s_wait_asynccnt(0);
#else
  asm volatile("s_wait_asynccnt 0" ::: "memory");
#endif
}

__global__ __launch_bounds__(NTHREADS)
void cheb_dist_conv2d_kernel(const float* __restrict__ x,
                             const float* __restrict__ w,
                             const float* __restrict__ bias,
                             float* __restrict__ out) {
  // LDS staging: haloed input tile + transposed weights
  __shared__ float xs[CIN][XTH][XTW];   // 16*10*18*4 = 11520 B
  __shared__ float wT[D_][COUT];        // 144*32*4  = 18432 B  (wT[d][co])

  const int tid = threadIdx.x;          // 0..127
  const int tx  = tid & (TW - 1);       // 0..15
  const int ty  = tid >> 4;             // 0..7
  const int bx0 = blockIdx.x * TW;      // tile origin x
  const int by0 = blockIdx.y * TH;      // tile origin y
  const int b   = blockIdx.z;

  // ---- stage weights, transposed, via async copies ----
  for (int i = tid; i < D_ * COUT; i += NTHREADS) {
    const int co = i / D_;
    const int d  = i - co * D_;
    async_ld_b32(&w[co * D_ + d], &wT[d][co]);
  }

  // ---- stage haloed x tile with replicate (clamp) padding ----
  const float* xb = x + (size_t)b * CIN * HH * WW;
  for (int i = tid; i < CIN * XTH * XTW; i += NTHREADS) {
    const int ci = i / (XTH * XTW);
    const int rem = i - ci * (XTH * XTW);
    const int r  = rem / XTW;
    const int c  = rem - r * XTW;
    int gy = by0 - 1 + r; gy = gy < 0 ? 0 : (gy > HH - 1 ? HH - 1 : gy);
    int gx = bx0 - 1 + c; gx = gx < 0 ? 0 : (gx > WW - 1 ? WW - 1 : gx);
    async_ld_b32(&xb[(ci * HH + gy) * WW + gx], &xs[ci][r][c]);
  }

  wait_async0();     // drain this wave's ASYNCcnt
  __syncthreads();   // make all waves' LDS writes visible

  // ---- Chebyshev distance: 32 accumulators per pixel, all in VGPRs ----
  float acc[COUT];
#pragma unroll
  for (int co = 0; co < COUT; ++co) acc[co] = 0.0f;  // |diff| >= 0

  for (int ci = 0; ci < CIN; ++ci) {
#pragma unroll
    for (int kh = 0; kh < KK; ++kh) {
#pragma unroll
      for (int kw = 0; kw < KK; ++kw) {
        const float xv = xs[ci][ty + kh][tx + kw];         // per-lane ds_load_b32
        const int d = (ci * KK + kh) * KK + kw;
        const float4* wr = (const float4*)(&wT[d][0]);     // uniform ds_load_b128
#pragma unroll
        for (int q = 0; q < COUT / 4; ++q) {
          const float4 wv = wr[q];
          acc[4 * q + 0] = fmaxf(acc[4 * q + 0], fabsf(xv - wv.x));
          acc[4 * q + 1] = fmaxf(acc[4 * q + 1], fabsf(xv - wv.y));
          acc[4 * q + 2] = fmaxf(acc[4 * q + 2], fabsf(xv - wv.z));
          acc[4 * q + 3] = fmaxf(acc[4 * q + 3], fabsf(xv - wv.w));
        }
      }
    }
  }

  // ---- write out + bias (NCHW), coalesced per co-plane ----
  const int gy = by0 + ty;
  const int gx = bx0 + tx;
#pragma unroll
  for (int co = 0; co < COUT; ++co) {
    out[(((size_t)b * COUT + co) * HH + gy) * WW + gx] = acc[co] + bias[co];
  }
}

extern "C" void kernel_launch(void* const* d_in, const int* in_sizes, int n_in,
                              void* d_out, int out_size, void* d_ws, size_t ws_size,
                              hipStream_t stream) {
  (void)in_sizes; (void)n_in; (void)d_ws; (void)ws_size; (void)out_size;
  const float* x    = (const float*)d_in[0];  // (8,16,64,64)
  const float* w    = (const float*)d_in[1];  // (32,16,3,3)
  const float* bias = (const float*)d_in[2];  // (32,1,1)
  float* out        = (float*)d_out;          // (8,32,64,64)

  dim3 grid(WW / TW, HH / TH, B_);            // (4, 8, 8) = 256 workgroups
  cheb_dist_conv2d_kernel<<<grid, NTHREADS, 0, stream>>>(x, w, bias, out);
}